// DeformSpaceAttention_26079041421808
// MI455X (gfx1250) — compile-verified
//
#include <hip/hip_runtime.h>

typedef float v2f __attribute__((ext_vector_type(2)));
typedef float v8f __attribute__((ext_vector_type(8)));

#define HDIM 100
#define WDIM 100
#define SPIX (HDIM * WDIM)
#define CCH 256
#define BATCH 4

// ---------------------------------------------------------------------------
// Kernel 1: transpose x [B, C, S] -> xT [B, S, C]  (channels-last for gathers)
// ---------------------------------------------------------------------------
__global__ __launch_bounds__(256) void transpose_kernel(const float* __restrict__ x,
                                                        float* __restrict__ xT) {
  __shared__ float tile[32][33];
  const int b  = blockIdx.z;
  const int c0 = blockIdx.y * 32;
  const int s0 = blockIdx.x * 32;
  const int tx = threadIdx.x;  // 0..31
  const int ty = threadIdx.y;  // 0..7

  const float* src = x + (size_t)b * CCH * SPIX;
#pragma unroll
  for (int i = 0; i < 4; ++i) {
    const int c = c0 + ty + i * 8;
    const int s = s0 + tx;
    float v = 0.0f;
    if (s < SPIX) v = src[(size_t)c * SPIX + s];
    tile[ty + i * 8][tx] = v;
  }
  __syncthreads();
  float* dst = xT + (size_t)b * SPIX * CCH;
#pragma unroll
  for (int i = 0; i < 4; ++i) {
    const int s = s0 + ty + i * 8;
    const int c = c0 + tx;
    if (s < SPIX) dst[(size_t)s * CCH + c] = tile[tx][ty + i * 8];
  }
}

// ---------------------------------------------------------------------------
// Kernel 2: deformable bilinear sample + 9-tap max + WMMA channel contraction
// Block = 256 threads = 8 wave32s, 16 pixels per block (2 pixels per wave).
// ---------------------------------------------------------------------------
__global__ __launch_bounds__(256) void deform_kernel(const float* __restrict__ xT,
                                                     const float* __restrict__ off,
                                                     const float* __restrict__ w0,
                                                     const float* __restrict__ b0,
                                                     float* __restrict__ out) {
  __shared__ float feat[16][260];  // stride 260: 16B-aligned rows, no bank conflicts

  const int lane = threadIdx.x & 31;
  const int wv   = threadIdx.x >> 5;
  const int blockPix = blockIdx.x * 16;

  // ---------------- Phase 1: per-pixel feat[C] via bilinear gather + max ----
  for (int pp = 0; pp < 2; ++pp) {
    const int p   = wv * 2 + pp;
    const int pix = blockPix + p;
    const int b   = pix / SPIX;
    const int rem = pix - b * SPIX;
    const int h   = rem / WDIM;
    const int wc  = rem - h * WDIM;

    const float* offp = off + (size_t)b * 18 * SPIX + (size_t)h * WDIM + wc;
    const float* xb   = xT + (size_t)b * SPIX * CCH;

    const int c1 = 4 * lane;        // channels [c1, c1+4)
    const int c2 = 128 + 4 * lane;  // channels [c2, c2+4)

    const float NEG = -__builtin_inff();
    float4 m1 = make_float4(NEG, NEG, NEG, NEG);
    float4 m2 = make_float4(NEG, NEG, NEG, NEG);

#pragma unroll
    for (int t = 0; t < 9; ++t) {
      const float dy = offp[(size_t)(2 * t) * SPIX];
      const float dx = offp[(size_t)(2 * t + 1) * SPIX];
      const float py = (float)h + (float)(t / 3 - 1) + dy;
      const float px = (float)wc + (float)(t % 3 - 1) + dx;
      const float y0f = floorf(py);
      const float x0f = floorf(px);
      const float wy = py - y0f;
      const float wx = px - x0f;
      const int y0 = (int)y0f, x0 = (int)x0f;
      const int y1 = y0 + 1, x1 = x0 + 1;

      const float vy0 = (y0 >= 0 && y0 < HDIM) ? 1.0f : 0.0f;
      const float vy1 = (y1 >= 0 && y1 < HDIM) ? 1.0f : 0.0f;
      const float vx0 = (x0 >= 0 && x0 < WDIM) ? 1.0f : 0.0f;
      const float vx1 = (x1 >= 0 && x1 < WDIM) ? 1.0f : 0.0f;

      const int yc0 = min(max(y0, 0), HDIM - 1);
      const int yc1 = min(max(y1, 0), HDIM - 1);
      const int xc0 = min(max(x0, 0), WDIM - 1);
      const int xc1 = min(max(x1, 0), WDIM - 1);

      const float w00 = (1.0f - wy) * (1.0f - wx) * vy0 * vx0;
      const float w01 = (1.0f - wy) * wx * vy0 * vx1;
      const float w10 = wy * (1.0f - wx) * vy1 * vx0;
      const float w11 = wy * wx * vy1 * vx1;

      const float* p00 = xb + ((size_t)yc0 * WDIM + xc0) * CCH;
      const float* p01 = xb + ((size_t)yc0 * WDIM + xc1) * CCH;
      const float* p10 = xb + ((size_t)yc1 * WDIM + xc0) * CCH;
      const float* p11 = xb + ((size_t)yc1 * WDIM + xc1) * CCH;

      {  // channel group 1 (coalesced 128b loads: lanes cover 128 contiguous ch)
        const float4 a = *reinterpret_cast<const float4*>(p00 + c1);
        const float4 bq = *reinterpret_cast<const float4*>(p01 + c1);
        const float4 cq = *reinterpret_cast<const float4*>(p10 + c1);
        const float4 dq = *reinterpret_cast<const float4*>(p11 + c1);
        m1.x = fmaxf(m1.x, fmaf(w00, a.x, fmaf(w01, bq.x, fmaf(w10, cq.x, w11 * dq.x))));
        m1.y = fmaxf(m1.y, fmaf(w00, a.y, fmaf(w01, bq.y, fmaf(w10, cq.y, w11 * dq.y))));
        m1.z = fmaxf(m1.z, fmaf(w00, a.z, fmaf(w01, bq.z, fmaf(w10, cq.z, w11 * dq.z))));
        m1.w = fmaxf(m1.w, fmaf(w00, a.w, fmaf(w01, bq.w, fmaf(w10, cq.w, w11 * dq.w))));
      }
      {  // channel group 2
        const float4 a = *reinterpret_cast<const float4*>(p00 + c2);
        const float4 bq = *reinterpret_cast<const float4*>(p01 + c2);
        const float4 cq = *reinterpret_cast<const float4*>(p10 + c2);
        const float4 dq = *reinterpret_cast<const float4*>(p11 + c2);
        m2.x = fmaxf(m2.x, fmaf(w00, a.x, fmaf(w01, bq.x, fmaf(w10, cq.x, w11 * dq.x))));
        m2.y = fmaxf(m2.y, fmaf(w00, a.y, fmaf(w01, bq.y, fmaf(w10, cq.y, w11 * dq.y))));
        m2.z = fmaxf(m2.z, fmaf(w00, a.z, fmaf(w01, bq.z, fmaf(w10, cq.z, w11 * dq.z))));
        m2.w = fmaxf(m2.w, fmaf(w00, a.w, fmaf(w01, bq.w, fmaf(w10, cq.w, w11 * dq.w))));
      }
    }
    *reinterpret_cast<float4*>(&feat[p][c1]) = m1;
    *reinterpret_cast<float4*>(&feat[p][c2]) = m2;
  }

  __syncthreads();

  // ---------------- Phase 2: 16-pixel x 256-channel dot via WMMA f32 16x16x4
  if (wv == 0) {  // full wave32, EXEC all ones at each wmma
    v8f acc = {0.f, 0.f, 0.f, 0.f, 0.f, 0.f, 0.f, 0.f};
    const int row  = lane & 15;        // A: M index
    const int koff = (lane >> 4) * 2;  // lanes 16-31 hold K=2,3
    // B column-0 mask: only N==0 (lanes 0 and 16) carries w0; branchless.
    const float sel = ((lane & 15) == 0) ? 1.0f : 0.0f;
    const float biasv = b0[0];
    const float* w0l = w0 + koff;  // per-lane-half base

    for (int k0 = 0; k0 < CCH; k0 += 4) {
      v2f a;
      a.x = feat[row][k0 + koff];
      a.y = feat[row][k0 + koff + 1];
      const float2 wvv = *reinterpret_cast<const float2*>(w0l + k0);
      v2f bm;
      bm.x = sel * wvv.x;
      bm.y = sel * wvv.y;
      acc = __builtin_amdgcn_wmma_f32_16x16x4_f32(false, a, false, bm,
                                                  (short)0, acc, false, false);
    }

    // D layout: lane 0 holds (M=0..7, N=0) in acc[0..7]; lane 16 holds M=8..15
    if (lane == 0 || lane == 16) {
      const int mb = (lane == 16) ? 8 : 0;
#pragma unroll
      for (int j = 0; j < 8; ++j) {
        const float z = acc[j] + biasv;
        out[blockPix + mb + j] = 1.0f / (1.0f + expf(-z));
      }
    }
  }
}

// ---------------------------------------------------------------------------
extern "C" void kernel_launch(void* const* d_in, const int* in_sizes, int n_in,
                              void* d_out, int out_size, void* d_ws, size_t ws_size,
                              hipStream_t stream) {
  const float* x   = (const float*)d_in[0];  // [4,256,100,100]
  const float* off = (const float*)d_in[1];  // [4,18,100,100]
  const float* w0  = (const float*)d_in[2];  // [1,256]
  const float* b0  = (const float*)d_in[3];  // [1]
  float* out = (float*)d_out;                // [4,1,100,100] = 40000
  float* xT  = (float*)d_ws;                 // 4*10000*256 floats = 40.96 MB

  dim3 tb(32, 8, 1);
  dim3 tg((SPIX + 31) / 32, CCH / 32, BATCH);
  transpose_kernel<<<tg, tb, 0, stream>>>(x, xT);

  const int nblocks = (BATCH * SPIX) / 16;  // 2500
  deform_kernel<<<nblocks, 256, 0, stream>>>(xT, off, w0, b0, out);
}